// BoundaryLoss_52553219834429
// MI455X (gfx1250) — compile-verified
//
#include <hip/hip_runtime.h>
#include <hip/hip_bf16.h>
#include <math.h>

#define BB   64
#define HH   128
#define WW   128
#define NPIX (BB*HH*WW)        // 1048576 = 2^20
#define NROWS (BB*HH)          // 8192 partials
#define BIGV 1000000.0f
#define THETA 0.7f

typedef __attribute__((ext_vector_type(2))) float v2f;
typedef __attribute__((ext_vector_type(8))) float v8f;

// ---------------------------------------------------------------------------
// Kernel 1: fused boundary mask + vertical EDT scan, one thread per column.
// boundary = (3x3 max > 3x3 min) over valid cells (SAME padding w/ +-inf).
// g[h] = min(fwd_dist, bwd_dist, BIG), scans start at BIG, reset to 0 on hit.
// ---------------------------------------------------------------------------
__global__ void colscan_kernel(const float* __restrict__ tgt, float* __restrict__ g) {
    int col = blockIdx.x * blockDim.x + threadIdx.x;   // b*WW + w
    if (col >= BB * WW) return;
    int b = col / WW, w = col % WW;
    const float* m = tgt + (size_t)b * HH * WW;

    unsigned bmask[HH / 32];
#pragma unroll
    for (int i = 0; i < HH / 32; ++i) bmask[i] = 0u;

    int w0 = (w > 0) ? w - 1 : 0;
    int w1 = (w < WW - 1) ? w + 1 : WW - 1;
    for (int h = 0; h < HH; ++h) {
        int h0 = (h > 0) ? h - 1 : 0;
        int h1 = (h < HH - 1) ? h + 1 : HH - 1;
        float mx = -3.0e38f, mn = 3.0e38f;
        for (int hh = h0; hh <= h1; ++hh)
            for (int ww = w0; ww <= w1; ++ww) {
                float v = m[hh * WW + ww];
                mx = fmaxf(mx, v);
                mn = fminf(mn, v);
            }
        if (mx > mn) bmask[h >> 5] |= (1u << (h & 31));
    }

    float* gc = g + (size_t)b * HH * WW + w;
    float cnt = BIGV;
    for (int h = 0; h < HH; ++h) {
        bool bd = (bmask[h >> 5] >> (h & 31)) & 1u;
        cnt = bd ? 0.0f : cnt + 1.0f;
        gc[h * WW] = cnt;
    }
    cnt = BIGV;
    for (int h = HH - 1; h >= 0; --h) {
        bool bd = (bmask[h >> 5] >> (h & 31)) & 1u;
        cnt = bd ? 0.0f : cnt + 1.0f;
        float fv = gc[h * WW];
        gc[h * WW] = fminf(fminf(fv, cnt), BIGV);
    }
}

// ---------------------------------------------------------------------------
// Kernel 2: one wave (32 threads) per row. Horizontal exact EDT min-plus,
// weight = exp(-theta*dist), stable BCE, then a WMMA-based f32 reduction of
// the 128 per-row products (ones-A 16x16x4 f32 => column sums in C/D).
// All 32 lanes active => EXEC all-ones at the WMMA (ISA requirement).
// ---------------------------------------------------------------------------
__global__ void row_kernel(const float* __restrict__ g,
                           const float* __restrict__ pred,
                           const float* __restrict__ tgt,
                           float* __restrict__ partials) {
    __shared__ float sq[WW];
    int row  = blockIdx.x;        // b*HH + h
    int lane = threadIdx.x;       // 0..31

    const float* grow = g + (size_t)row * WW;
#pragma unroll
    for (int k = 0; k < WW / 32; ++k) {
        float gv = grow[lane + k * 32];
        sq[lane + k * 32] = gv * gv;
    }
    __syncthreads();

    float prod[4];
#pragma unroll
    for (int k = 0; k < 4; ++k) {
        int i = lane + k * 32;
        float best = 3.0e38f;
        for (int j = 0; j < WW; ++j) {
            int d = i - j;
            best = fminf(best, sq[j] + (float)(d * d));
        }
        float dist = sqrtf(best);
        float wgt  = expf(-THETA * dist);
        size_t idx = (size_t)row * WW + (size_t)i;
        float x = pred[idx];
        float t = tgt[idx];
        float sp = fmaxf(x, 0.0f) + log1pf(expf(-fabsf(x)));  // logaddexp(0,x)
        prod[k] = wgt * (sp - x * t);
    }

    // Pairwise pre-add -> 64 values across the wave -> one f32 WMMA reduction.
    v2f a;  a.x = 1.0f;              a.y = 1.0f;
    v2f bv; bv.x = prod[0] + prod[1]; bv.y = prod[2] + prod[3];
    v8f c = {};
    c = __builtin_amdgcn_wmma_f32_16x16x4_f32(false, a, false, bv,
                                              (short)0, c, false, false);
    float s = 0.0f;
#pragma unroll
    for (int i = 0; i < 8; ++i) s += c[i];
#pragma unroll
    for (int off = 16; off >= 1; off >>= 1)
        s += __shfl_xor(s, off, 32);
    if (lane == 0) partials[row] = s * (1.0f / 16.0f);  // rows of D duplicate 16x
}

// ---------------------------------------------------------------------------
// Kernel 3: single block, 8 waves. Each wave WMMA-accumulates 1024 partials
// (16 iterations of K=64) in full f32, then cross-lane + LDS combine.
// Deterministic (no float atomics).
// ---------------------------------------------------------------------------
__global__ void reduce_kernel(const float* __restrict__ partials,
                              float* __restrict__ out) {
    __shared__ float sdata[8];
    int tid  = threadIdx.x;
    int wave = tid >> 5, lane = tid & 31;

    v2f a; a.x = 1.0f; a.y = 1.0f;
    v8f c = {};
    int base = wave * 1024;                 // 8 waves * 1024 = 8192 = NROWS
    for (int it = 0; it < 16; ++it) {
        v2f bv;
        bv.x = partials[base + it * 64 + lane];
        bv.y = partials[base + it * 64 + 32 + lane];
        c = __builtin_amdgcn_wmma_f32_16x16x4_f32(false, a, false, bv,
                                                  (short)0, c, false, false);
    }
    float s = 0.0f;
#pragma unroll
    for (int i = 0; i < 8; ++i) s += c[i];
#pragma unroll
    for (int off = 16; off >= 1; off >>= 1)
        s += __shfl_xor(s, off, 32);
    if (lane == 0) sdata[wave] = s * (1.0f / 16.0f);
    __syncthreads();
    if (tid == 0) {
        float tot = 0.0f;
#pragma unroll
        for (int i = 0; i < 8; ++i) tot += sdata[i];
        out[0] = tot * (1.0f / (float)NPIX);   // mean, /2^20 exact
    }
}

// ---------------------------------------------------------------------------
extern "C" void kernel_launch(void* const* d_in, const int* in_sizes, int n_in,
                              void* d_out, int out_size, void* d_ws, size_t ws_size,
                              hipStream_t stream) {
    (void)in_sizes; (void)n_in; (void)out_size; (void)ws_size;
    const float* pred = (const float*)d_in[0];   // [64,1,128,128] f32
    const float* tgt  = (const float*)d_in[1];   // [64,1,128,128] f32 binary

    float* g        = (float*)d_ws;              // NPIX floats (4 MB)
    float* partials = g + NPIX;                  // NROWS floats (32 KB)

    colscan_kernel<<<(BB * WW + 255) / 256, 256, 0, stream>>>(tgt, g);
    row_kernel<<<NROWS, 32, 0, stream>>>(g, pred, tgt, partials);
    reduce_kernel<<<1, 256, 0, stream>>>(partials, (float*)d_out);
}